// HebbianNet_62569083568469
// MI455X (gfx1250) — compile-verified
//
#include <hip/hip_runtime.h>
#include <hip/hip_bf16.h>

typedef __attribute__((ext_vector_type(2))) float v2f;
typedef __attribute__((ext_vector_type(8))) float v8f;

// ---------------------------------------------------------------------------
// Phase 1: vj = relu(x @ w^T + bias)
//   x: (8, I) fp32, w: (O, I) fp32, vj: (8, O) fp32
// One wave per 16-wide N tile, M padded 8->16 (rows 8..15 duplicate row 7 and
// are discarded). Uses V_WMMA_F32_16X16X4_F32 (full fp32 precision).
//
// Fragment layout per CDNA5 ISA 7.12.2 (wave32):
//   A 16x4 f32, 2 VGPRs: lanes 0-15 -> M=lane, {v0,v1} = K=0,1
//                        lanes16-31 -> M=lane-16, {v0,v1} = K=2,3
//   B 4x16 f32, 2 VGPRs: lanes 0-15 -> N=lane, {v0,v1} = K=0,1
//                        lanes16-31 -> N=lane-16, {v0,v1} = K=2,3
//   C/D 16x16 f32, 8 VGPRs: lanes 0-15 -> N=lane, v[r] = M=r  (M=0..7 here)
// ---------------------------------------------------------------------------
__global__ __launch_bounds__(128) void gemm_relu_wmma(
    const float* __restrict__ x, const float* __restrict__ w,
    const float* __restrict__ bias, float* __restrict__ vj,
    int I, int O)
{
    const int wave = (blockIdx.x * blockDim.x + threadIdx.x) >> 5;
    const int lane = threadIdx.x & 31;
    const int n0   = wave * 16;
    if (n0 >= O) return;

    const int half = lane >> 4;   // which K pair this lane supplies
    const int lid  = lane & 15;   // M for A-frag, N for B-frag
    const int m    = lid < 8 ? lid : 7;          // pad rows 8..15 (discarded)

    const float* __restrict__ xrow = x + (size_t)m * I + half * 2;
    const float* __restrict__ wrow = w + (size_t)(n0 + lid) * I + half * 2;

    v8f c = {};
    for (int k = 0; k < I; k += 4) {
        v2f a = *(const v2f*)(xrow + k);   // x[m, k+2h], x[m, k+2h+1]
        v2f b = *(const v2f*)(wrow + k);   // w[n, k+2h], w[n, k+2h+1]
        c = __builtin_amdgcn_wmma_f32_16x16x4_f32(
                false, a, false, b, (short)0, c, false, false);
    }

    if (half == 0) {                       // lanes 0-15 hold M=0..7
        const float bb = bias[n0 + lid];
        #pragma unroll
        for (int r = 0; r < 8; ++r) {
            float v = c[r] + bb;
            vj[(size_t)r * O + n0 + lid] = v > 0.f ? v : 0.f;
        }
    }
}

// ---------------------------------------------------------------------------
// Phase 2: out[b,o] = vj[b,o] + rate*( sum_i vi[b,i] * s(b,o,i)
//                                      + c2b * sum_i vi[b,i] )
//   s = sum_k c2w[k]*relu(alpha_k*vi + beta_k*w[o,i] + gamma_k*vj[b,o]+c1b[k])
// One block per o; 8 waves = 8 batch rows; w row staged in LDS; i-loop
// unrolled x4 with float4 loads; wave32 shuffle reduction.
// ---------------------------------------------------------------------------
__global__ __launch_bounds__(256) void hebbian_update(
    const float* __restrict__ vi,   // (8, I)
    const float* __restrict__ w,    // (O, I)
    const float* __restrict__ vj,   // (8, O)
    const float* __restrict__ c1w,  // (8, 3) [alpha, beta, gamma]
    const float* __restrict__ c1b,  // (8,)
    const float* __restrict__ c2w,  // (8,)
    const float* __restrict__ c2b,  // (1,)
    const int*   __restrict__ batch_num, // (1,)
    float* __restrict__ out,        // (8, O)
    int I, int O)
{
    extern __shared__ float lds_w[];           // I floats
    const int o    = blockIdx.x;
    const int tid  = threadIdx.x;
    const int lane = tid & 31;
    const int b    = tid >> 5;                 // wave index == batch row (8)

    // stage w[o, :] into LDS (coalesced float4)
    const float* __restrict__ wrow = w + (size_t)o * I;
    for (int i = tid * 4; i < I; i += 256 * 4)
        *(float4*)(lds_w + i) = *(const float4*)(wrow + i);
    __syncthreads();

    const float vjv = vj[(size_t)b * O + o];
    float alpha[8], beta[8], d[8], cw[8];
    #pragma unroll
    for (int k = 0; k < 8; ++k) {
        alpha[k] = c1w[k * 3 + 0];
        beta[k]  = c1w[k * 3 + 1];
        d[k]     = __builtin_fmaf(c1w[k * 3 + 2], vjv, c1b[k]);
        cw[k]    = c2w[k];
    }

    const float* __restrict__ virow = vi + (size_t)b * I;
    float acc0 = 0.f, acc1 = 0.f, acc2 = 0.f, acc3 = 0.f, sv = 0.f;
    for (int i = lane * 4; i < I; i += 128) {
        float4 a4 = *(const float4*)(virow + i);
        float4 w4 = *(const float4*)(lds_w + i);
        sv += (a4.x + a4.y) + (a4.z + a4.w);
        float s0 = 0.f, s1 = 0.f, s2 = 0.f, s3 = 0.f;
        #pragma unroll
        for (int k = 0; k < 8; ++k) {
            float t0 = __builtin_fmaf(beta[k], w4.x, __builtin_fmaf(alpha[k], a4.x, d[k]));
            float t1 = __builtin_fmaf(beta[k], w4.y, __builtin_fmaf(alpha[k], a4.y, d[k]));
            float t2 = __builtin_fmaf(beta[k], w4.z, __builtin_fmaf(alpha[k], a4.z, d[k]));
            float t3 = __builtin_fmaf(beta[k], w4.w, __builtin_fmaf(alpha[k], a4.w, d[k]));
            t0 = t0 > 0.f ? t0 : 0.f;
            t1 = t1 > 0.f ? t1 : 0.f;
            t2 = t2 > 0.f ? t2 : 0.f;
            t3 = t3 > 0.f ? t3 : 0.f;
            s0 = __builtin_fmaf(cw[k], t0, s0);
            s1 = __builtin_fmaf(cw[k], t1, s1);
            s2 = __builtin_fmaf(cw[k], t2, s2);
            s3 = __builtin_fmaf(cw[k], t3, s3);
        }
        acc0 = __builtin_fmaf(a4.x, s0, acc0);
        acc1 = __builtin_fmaf(a4.y, s1, acc1);
        acc2 = __builtin_fmaf(a4.z, s2, acc2);
        acc3 = __builtin_fmaf(a4.w, s3, acc3);
    }
    float acc = (acc0 + acc1) + (acc2 + acc3);

    // wave32 reduction
    #pragma unroll
    for (int off = 16; off > 0; off >>= 1) {
        acc += __shfl_down(acc, off, 32);
        sv  += __shfl_down(sv,  off, 32);
    }

    if (lane == 0) {
        const float rate = 0.001f / (float)batch_num[0];
        out[(size_t)b * O + o] = vjv + rate * (acc + c2b[0] * sv);
    }
}

// ---------------------------------------------------------------------------
extern "C" void kernel_launch(void* const* d_in, const int* in_sizes, int n_in,
                              void* d_out, int out_size, void* d_ws, size_t ws_size,
                              hipStream_t stream)
{
    const float* x   = (const float*)d_in[0];   // (8, 1024)
    const float* w1  = (const float*)d_in[1];   // (2048, 1024)
    const float* b1  = (const float*)d_in[2];   // (2048,)
    const float* w2  = (const float*)d_in[3];   // (512, 2048)
    const float* b2  = (const float*)d_in[4];   // (512,)
    const float* c1w = (const float*)d_in[5];   // (8, 3)
    const float* c1b = (const float*)d_in[6];   // (8,)
    const float* c2w = (const float*)d_in[7];   // (8,)
    const float* c2b = (const float*)d_in[8];   // (1,)
    const int*   bn  = (const int*)  d_in[9];   // (1,)

    const int B = 8, I1 = 1024, O1 = 2048, O2 = 512;

    float* ws   = (float*)d_ws;
    float* vj1  = ws;                     // 8*2048
    float* out1 = ws + (size_t)B * O1;    // 8*2048
    float* vj2  = out1 + (size_t)B * O1;  // 8*512
    float* out2 = (float*)d_out;          // 8*512

    // Layer 1
    gemm_relu_wmma<<<O1 / 64, 128, 0, stream>>>(x, w1, b1, vj1, I1, O1);
    hebbian_update<<<O1, 256, I1 * sizeof(float), stream>>>(
        x, w1, vj1, c1w, c1b, c2w, c2b, bn, out1, I1, O1);

    // Layer 2 (I = O1)
    gemm_relu_wmma<<<O2 / 64, 128, 0, stream>>>(out1, w2, b2, vj2, O1, O2);
    hebbian_update<<<O2, 256, O1 * sizeof(float), stream>>>(
        out1, w2, vj2, c1w, c1b, c2w, c2b, bn, out2, O1, O2);
}